// RUNG_learnable_combined_31104153157799
// MI455X (gfx1250) — compile-verified
//
#include <hip/hip_runtime.h>
#include <math.h>

#define N_NODES 4096
#define IN_DIM  512
#define HID     256
#define OUT_DIM 64
#define PROP_STEPS 10
#define CAP 128                       // max neighbors per row (E[deg]~41, +8 sigma safe)
#define LAMV ((1.0f/0.9f) - 1.0f)     // lam = 1/lam_hat - 1
#define SCADA 3.7f
#define EPSV 1e-8f

typedef float v2f __attribute__((ext_vector_type(2)));
typedef float v8f __attribute__((ext_vector_type(8)));

// ---------------------------------------------------------------------------
// 1) Dense A (0/1, zero diag) -> CSR (sorted cols), D = rowsum + 1, D^-1/2.
//    One wave per row; ballot/popcount compaction => deterministic order.
// ---------------------------------------------------------------------------
__global__ void build_csr_kernel(const float* __restrict__ A,
                                 int* __restrict__ cols, int* __restrict__ cnt,
                                 float* __restrict__ Dv, float* __restrict__ rsqD) {
  int wave = (blockIdx.x * blockDim.x + threadIdx.x) >> 5;
  int lane = threadIdx.x & 31;
  if (wave >= N_NODES) return;
  const int row = wave;
  const float* __restrict__ arow = A + (size_t)row * N_NODES;
  int count = 0;
  float dsum = 0.0f;
  for (int base = 0; base < N_NODES; base += 32) {
    float a = arow[base + lane];
    dsum += a;
    bool nz = (a != 0.0f) && (base + lane != row);
    unsigned mask = (unsigned)__ballot(nz);            // wave32: low 32 bits
    int off = __popc(mask & ((1u << lane) - 1u));
    if (nz && (count + off) < CAP)
      cols[(size_t)row * CAP + count + off] = base + lane;
    count += __popc(mask);
  }
  for (int o = 16; o > 0; o >>= 1) dsum += __shfl_xor(dsum, o, 32);
  if (lane == 0) {
    cnt[row] = (count > CAP) ? CAP : count;
    float d = dsum + 1.0f;                             // add_loops: +I
    Dv[row] = d;
    rsqD[row] = rsqrtf(d);
  }
}

// ---------------------------------------------------------------------------
// 2) fp32 GEMM via V_WMMA_F32_16X16X4_F32: C = act(A[M,K]@B[K,N] + bias).
//    Register-blocked: each wave owns a 16(M) x 64(N) strip -> one A-frag
//    load feeds 4 WMMA per K-step. 4 waves/block cover 64 rows.
// ---------------------------------------------------------------------------
__global__ void wmma_gemm_bias_kernel(const float* __restrict__ A,
                                      const float* __restrict__ B,
                                      const float* __restrict__ bias,
                                      float* __restrict__ C,
                                      int M, int K, int N, int do_relu) {
  const int wave = threadIdx.x >> 5;
  const int lane = threadIdx.x & 31;
  const int m0 = blockIdx.x * 64 + wave * 16;
  const int n0 = blockIdx.y * 64;
  if (m0 >= M) return;                                 // wave-uniform: EXEC stays full
  const bool hi = (lane >= 16);
  const int l15 = lane & 15;
  const int kb = hi ? 2 : 0;                           // A/B frag K sub-offset
  const int mrow = m0 + l15;
  const float* __restrict__ arow = A + (size_t)mrow * K + kb;

  v8f acc[4];
#pragma unroll
  for (int t = 0; t < 4; ++t) acc[t] = (v8f){0.f,0.f,0.f,0.f,0.f,0.f,0.f,0.f};

  for (int k = 0; k < K; k += 4) {
    if (((k & 63) == 0) && (k + 128 < K)) {
      __builtin_prefetch(arow + k + 128, 0, 3);                        // global_prefetch_b8
      __builtin_prefetch(&B[(size_t)(k + 128 + kb) * N + n0 + l15], 0, 3);
    }
    v2f a = *(const v2f*)(arow + k);                   // A[mrow][k+kb .. +1]
    const float* __restrict__ b0p = &B[(size_t)(k + kb) * N + n0 + l15];
    const float* __restrict__ b1p = &B[(size_t)(k + kb + 1) * N + n0 + l15];
#pragma unroll
    for (int t = 0; t < 4; ++t) {
      v2f b;
      b.x = b0p[t * 16];
      b.y = b1p[t * 16];
      acc[t] = __builtin_amdgcn_wmma_f32_16x16x4_f32(false, a, false, b,
                                                     (short)0, acc[t], false, false);
    }
  }
#pragma unroll
  for (int t = 0; t < 4; ++t) {
    const int ncol = n0 + t * 16 + l15;
    const float bv = bias[ncol];
#pragma unroll
    for (int v = 0; v < 8; ++v) {
      int row = m0 + v + (hi ? 8 : 0);
      float out = acc[t][v] + bv;
      if (do_relu) out = fmaxf(out, 0.0f);
      C[(size_t)row * N + ncol] = out;
    }
  }
}

// ---------------------------------------------------------------------------
// 3) Row normalize: Fu = (Fc * D^-1/2) / max(||.||, 1e-12). Wave per row,
//    lane owns channel pair {2l, 2l+1} -> single b64 load/store per lane.
// ---------------------------------------------------------------------------
__global__ void normalize_kernel(const float* __restrict__ Fc,
                                 const float* __restrict__ rsqD,
                                 float* __restrict__ Fu) {
  int wave = (blockIdx.x * blockDim.x + threadIdx.x) >> 5;
  int lane = threadIdx.x & 31;
  if (wave >= N_NODES) return;
  const int row = wave;
  const float rs = rsqD[row];
  v2f f = *(const v2f*)&Fc[(size_t)row * OUT_DIM + 2 * lane];
  f.x *= rs; f.y *= rs;
  float s = f.x * f.x + f.y * f.y;
  for (int o = 16; o > 0; o >>= 1) s += __shfl_xor(s, o, 32);
  const float inv = 1.0f / fmaxf(sqrtf(s), 1e-12f);
  v2f out; out.x = f.x * inv; out.y = f.y * inv;
  *(v2f*)&Fu[(size_t)row * OUT_DIM + 2 * lane] = out;
}

// ---------------------------------------------------------------------------
// 4) Sparse SCAD-weighted propagation. Wave per row; per edge: one b64 load
//    per operand + butterfly dot for cosine distance, SCAD weight, then
//    accumulate (W*A_tilde)@Fc and Q_hat.
// ---------------------------------------------------------------------------
__global__ void prop_kernel(const float* __restrict__ Fu, const float* __restrict__ Fc,
                            const float* __restrict__ F0, const int* __restrict__ cols,
                            const int* __restrict__ cnt, const float* __restrict__ Dv,
                            const float* __restrict__ rsqD,
                            const float* __restrict__ raw_gamma, int step,
                            float* __restrict__ Fn) {
  int wave = (blockIdx.x * blockDim.x + threadIdx.x) >> 5;
  int lane = threadIdx.x & 31;
  if (wave >= N_NODES) return;
  const int row = wave;
  const int c2 = 2 * lane;

  const float g = raw_gamma[step];
  const float gamma = 2.0f / (1.0f + expf(-g));        // 2*sigmoid
  const float lam_k = fmaxf(gamma / SCADA, EPSV);
  const float t = SCADA * lam_k;
  const float rs_i = rsqD[row];

  const v2f fu = *(const v2f*)&Fu[(size_t)row * OUT_DIM + c2];

  float accq = 0.0f;
  v2f acc; acc.x = 0.0f; acc.y = 0.0f;
  const int deg = cnt[row];
  const int* __restrict__ cr = cols + (size_t)row * CAP;
  for (int e = 0; e < deg; ++e) {
    const int j = cr[e];
    if (e + 1 < deg) {
      int jn = cr[e + 1];
      __builtin_prefetch(&Fu[(size_t)jn * OUT_DIM + c2], 0, 3);
      __builtin_prefetch(&Fc[(size_t)jn * OUT_DIM + c2], 0, 3);
    }
    const v2f gj = *(const v2f*)&Fu[(size_t)j * OUT_DIM + c2];
    float p = fu.x * gj.x + fu.y * gj.y;
    for (int o = 16; o > 0; o >>= 1) p += __shfl_xor(p, o, 32);
    const float y = 1.0f - p;                          // cosine distance
    float w;
    if (y <= lam_k) {
      w = 1.0f;
    } else if (y <= t) {
      float sy = (y <= 0.0f) ? 1.0f : y;
      w = (t - y) / ((SCADA - 1.0f) * sy);
    } else {
      w = 0.0f;
    }
    if (w != w) w = 1.0f;                              // NaN -> 1 (reference)
    accq += w;                                         // (W*A).sum: A_ij == 1
    const float coef = w * rs_i * rsqD[j];             // W * A_tilde
    const v2f fcj = *(const v2f*)&Fc[(size_t)j * OUT_DIM + c2];
    acc.x = fmaf(coef, fcj.x, acc.x);
    acc.y = fmaf(coef, fcj.y, acc.y);
  }
  const float invq = 1.0f / (accq / Dv[row] + LAMV);   // 1/Q_hat
  const v2f f0v = *(const v2f*)&F0[(size_t)row * OUT_DIM + c2];
  v2f res;
  res.x = fmaf(LAMV, f0v.x, acc.x) * invq;
  res.y = fmaf(LAMV, f0v.y, acc.y) * invq;
  *(v2f*)&Fn[(size_t)row * OUT_DIM + c2] = res;
}

// ---------------------------------------------------------------------------
extern "C" void kernel_launch(void* const* d_in, const int* in_sizes, int n_in,
                              void* d_out, int out_size, void* d_ws, size_t ws_size,
                              hipStream_t stream) {
  const float* A         = (const float*)d_in[0];
  const float* F         = (const float*)d_in[1];
  const float* W1        = (const float*)d_in[2];
  const float* b1        = (const float*)d_in[3];
  const float* W2        = (const float*)d_in[4];
  const float* b2        = (const float*)d_in[5];
  const float* raw_gamma = (const float*)d_in[6];
  float* out = (float*)d_out;

  char* ws = (char*)d_ws;
  size_t off = 0;
  auto alloc = [&](size_t bytes) -> void* {
    void* p = ws + off;
    off = (off + bytes + 255) & ~(size_t)255;
    return p;
  };
  int*   cols = (int*)  alloc((size_t)N_NODES * CAP * sizeof(int));
  int*   cnt  = (int*)  alloc((size_t)N_NODES * sizeof(int));
  float* Dv   = (float*)alloc((size_t)N_NODES * sizeof(float));
  float* rsq  = (float*)alloc((size_t)N_NODES * sizeof(float));
  float* H    = (float*)alloc((size_t)N_NODES * HID * sizeof(float));
  float* F0   = (float*)alloc((size_t)N_NODES * OUT_DIM * sizeof(float));
  float* Fu   = (float*)alloc((size_t)N_NODES * OUT_DIM * sizeof(float));
  float* P0   = (float*)alloc((size_t)N_NODES * OUT_DIM * sizeof(float));
  float* P1   = (float*)alloc((size_t)N_NODES * OUT_DIM * sizeof(float));

  // CSR + degrees from dense adjacency (single HBM pass over 67 MB)
  build_csr_kernel<<<N_NODES / 8, 256, 0, stream>>>(A, cols, cnt, Dv, rsq);

  // MLP: H = relu(F@W1 + b1); F0 = H@W2 + b2   (fp32 WMMA, 16x64 strip/wave)
  dim3 g1(N_NODES / 64, HID / 64);
  wmma_gemm_bias_kernel<<<g1, 128, 0, stream>>>(F, W1, b1, H, N_NODES, IN_DIM, HID, 1);
  dim3 g2(N_NODES / 64, OUT_DIM / 64);
  wmma_gemm_bias_kernel<<<g2, 128, 0, stream>>>(H, W2, b2, F0, N_NODES, HID, OUT_DIM, 0);

  // 10 propagation steps; last one writes d_out directly
  const float* src = F0;
  for (int k = 0; k < PROP_STEPS; ++k) {
    normalize_kernel<<<N_NODES / 8, 256, 0, stream>>>(src, rsq, Fu);
    float* dst = (k == PROP_STEPS - 1) ? out : ((k & 1) ? P1 : P0);
    prop_kernel<<<N_NODES / 8, 256, 0, stream>>>(Fu, src, F0, cols, cnt, Dv, rsq,
                                                 raw_gamma, k, dst);
    src = dst;
  }
}